// PointWiseBlockMaching_26800595927727
// MI455X (gfx1250) — compile-verified
//
#include <hip/hip_runtime.h>
#include <stdint.h>

// ---------------------------------------------------------------------------
// Point-wise block matching cost volume for MI455X (gfx1250).
//
//  * Integer-stride windows/disparities collapse bilinear sampling to one
//    contiguous 2D texel tile per (point, level, channel) + separable lerps
//    with a single shared (fx,fy) weight pair.
//  * Tiles staged global->LDS with the CDNA5 Tensor Data Mover
//    (tensor_load_to_lds / TENSORcnt), DOUBLE-BUFFERED and issued two
//    channels ahead so DMA overlaps both interpolation and SAD phases
//    (TENSORcnt completes in-order per wave -> s_wait_tensorcnt 4 retires
//    exactly the older channel's 4 tiles).
//  * One 64-thread block per (point, level-pair j): scale-1 level j and
//    scale-4 level j+8, fused 48->192 upsample + pair sum.
// ---------------------------------------------------------------------------

typedef uint32_t u32;
typedef __attribute__((ext_vector_type(4))) unsigned int uintv4;
typedef __attribute__((ext_vector_type(8))) int          intv8;
typedef __attribute__((ext_vector_type(4))) int          intv4;

#define NLEV  16
#define NCH   16
#define IMH   96
#define IMW   320
#define NPTS  1024
#define MAXD  192
#define ND4   48

// tile geometry maxima (pre-clip)
#define R1_W 197
#define R1_H 6
#define L1_W 6
#define L1_H 6
#define R4_W 206
#define R4_H 18
#define L4_W 18
#define L4_H 18

#if __has_builtin(__builtin_amdgcn_tensor_load_to_lds)
#define HAVE_TDM 1
#else
#define HAVE_TDM 0
#endif

struct Rect { int cs, cw, rs, rh; };

__device__ __forceinline__ Rect clip_rect(int clo, int chi, int rlo, int rhi) {
  Rect r;
  r.cs = clo > 0 ? clo : 0;
  int ce = chi < IMW - 1 ? chi : IMW - 1;
  r.cw = ce - r.cs + 1;
  r.rs = rlo > 0 ? rlo : 0;
  int re = rhi < IMH - 1 ? rhi : IMH - 1;
  r.rh = re - r.rs + 1;
  return r;
}

// low 32 bits of a generic LDS address == LDS byte offset (ISA 10.2 aperture)
__device__ __forceinline__ u32 lds_off_u32(const void* p) {
  return (u32)(uintptr_t)p;
}

#if HAVE_TDM
// Build a 2D-tile D# (ISA cdna5 §8.3/8.4) and issue TENSOR_LOAD_TO_LDS.
// tile == tensor (pre-clipped, always in bounds) so the OOB path is unused.
__device__ __forceinline__ void tdm_load_tile_2d(u32 lds_byte_off,
                                                 const float* gsrc,
                                                 int tile_w, int tile_h) {
  uint64_t ga = (uint64_t)(uintptr_t)gsrc;
  u32 tw = (u32)tile_w, th = (u32)tile_h;
  uintv4 g0;
  g0.x = 1u;                                         // count=1, user mode
  g0.y = lds_byte_off;                               // lds_addr (bytes)
  g0.z = (u32)ga;                                    // global_addr[31:0]
  g0.w = (u32)((ga >> 32) & 0x01ffffffull) | (2u << 30); // addr[56:32] | type=2
  intv8 g1;
  g1[0] = (int)(2u << 16);                           // data_size = 4B
  g1[1] = (int)((tw & 0xffffu) << 16);               // tensor_dim0[15:0]
  g1[2] = (int)(((tw >> 16) & 0xffffu) | ((th & 0xffffu) << 16)); // d0 hi | d1 lo
  g1[3] = (int)(((th >> 16) & 0xffffu) | ((tw & 0xffffu) << 16)); // d1 hi | tile_dim0
  g1[4] = (int)(th & 0xffffu);                       // tile_dim1 (tile_dim2=0 -> 2D)
  g1[5] = (int)(u32)IMW;                             // tensor_dim0_stride (row pitch)
  g1[6] = 0;
  g1[7] = 0;
  intv4 z4; z4[0] = 0; z4[1] = 0; z4[2] = 0; z4[3] = 0;
#if __clang_major__ >= 23
  intv8 z8; z8[0]=0; z8[1]=0; z8[2]=0; z8[3]=0; z8[4]=0; z8[5]=0; z8[6]=0; z8[7]=0;
  __builtin_amdgcn_tensor_load_to_lds(g0, g1, z4, z4, z8, 0);
#else
  __builtin_amdgcn_tensor_load_to_lds(g0, g1, z4, z4, 0);
#endif
}
#endif

__device__ __forceinline__ const float* tile_src(const float* base, int lev, int c,
                                                 const Rect& rc) {
  return base + (((size_t)(lev * NCH + c)) * IMH + rc.rs) * IMW + rc.cs;
}

#if !HAVE_TDM
__device__ __forceinline__ void copy_tile(float* dst, const float* base, int lev,
                                          int c, const Rect& rc, int tid) {
  const float* g = tile_src(base, lev, c, rc);
  for (int r = 0; r < rc.rh; ++r)
    for (int col = tid; col < rc.cw; col += 64)
      dst[r * rc.cw + col] = g[r * IMW + col];
}
#endif

__device__ __forceinline__ float texel(const float* s, const Rect& rc, int x, int y) {
  // zeros-padding: anything outside the clipped rect is outside the image
  if (x < rc.cs || x >= rc.cs + rc.cw || y < rc.rs || y >= rc.rs + rc.rh) return 0.0f;
  return s[(y - rc.rs) * rc.cw + (x - rc.cs)];
}

__device__ __forceinline__ float bilerp(const float* s, const Rect& rc,
                                        int x0, int y0, float fx, float fy) {
  float v00 = texel(s, rc, x0,     y0);
  float v01 = texel(s, rc, x0 + 1, y0);
  float v10 = texel(s, rc, x0,     y0 + 1);
  float v11 = texel(s, rc, x0 + 1, y0 + 1);
  float a = v00 + fx * (v01 - v00);
  float b = v10 + fx * (v11 - v10);
  return a + fy * (b - a);
}

extern "C" __global__ __launch_bounds__(64)
void PointWiseBlockMaching_26800595927727_kernel(const float* __restrict__ lfeat,
                                                 const float* __restrict__ rfeat,
                                                 const float* __restrict__ pts,
                                                 float* __restrict__ out)
{
  // double-buffered texel tiles (ping-pong across channels)
  __shared__ float sR1[2][R1_H * R1_W];
  __shared__ float sL1[2][L1_H * L1_W];
  __shared__ float sR4[2][R4_H * R4_W];
  __shared__ float sL4[2][L4_H * L4_W];
  __shared__ float rv1[5 * 196];   // right interp values, scale 1, k = idx-193
  __shared__ float rv4[5 * 52];    // right interp values, scale 4, k = idx-49
  __shared__ float lv1[25];
  __shared__ float lv4[25];
  __shared__ float c4s[ND4];

  const int tid  = (int)threadIdx.x;
  const int wid  = tid >> 5;
  const int blk  = (int)blockIdx.x;
  const int n    = blk >> 3;
  const int lev1 = blk & 7;        // scale-1 level
  const int lev4 = lev1 + 8;       // scale-4 level

  const float px = pts[2 * n + 0];
  const float py = pts[2 * n + 1];
  const int   ix0 = (int)floorf(px);
  const int   iy0 = (int)floorf(py);
  const float fxw = px - floorf(px);
  const float fyw = py - floorf(py);

  // texel footprints (k = wx-2-d): scale1 k in [-193,2]; scale4 k in [-49,2]
  const Rect rR1 = clip_rect(ix0 - 193, ix0 + 3, iy0 - 2, iy0 + 3);
  const Rect rL1 = clip_rect(ix0 - 2,   ix0 + 3, iy0 - 2, iy0 + 3);
  const Rect rR4 = clip_rect(ix0 - 196, ix0 + 9, iy0 - 8, iy0 + 9);
  const Rect rL4 = clip_rect(ix0 - 8,   ix0 + 9, iy0 - 8, iy0 + 9);

  float acc1[3] = {0.f, 0.f, 0.f};
  float acc4 = 0.f;

#if HAVE_TDM
  // prime the pipeline two channels deep (TENSORcnt completes in-order)
  if (wid == 0) {
    tdm_load_tile_2d(lds_off_u32(sR1[0]), tile_src(rfeat, lev1, 0, rR1), rR1.cw, rR1.rh);
    tdm_load_tile_2d(lds_off_u32(sL1[0]), tile_src(lfeat, lev1, 0, rL1), rL1.cw, rL1.rh);
    tdm_load_tile_2d(lds_off_u32(sR4[0]), tile_src(rfeat, lev4, 0, rR4), rR4.cw, rR4.rh);
    tdm_load_tile_2d(lds_off_u32(sL4[0]), tile_src(lfeat, lev4, 0, rL4), rL4.cw, rL4.rh);
    tdm_load_tile_2d(lds_off_u32(sR1[1]), tile_src(rfeat, lev1, 1, rR1), rR1.cw, rR1.rh);
    tdm_load_tile_2d(lds_off_u32(sL1[1]), tile_src(lfeat, lev1, 1, rL1), rL1.cw, rL1.rh);
    tdm_load_tile_2d(lds_off_u32(sR4[1]), tile_src(rfeat, lev4, 1, rR4), rR4.cw, rR4.rh);
    tdm_load_tile_2d(lds_off_u32(sL4[1]), tile_src(lfeat, lev4, 1, rL4), rL4.cw, rL4.rh);
  }
#endif

  for (int c = 0; c < NCH; ++c) {
    const int buf = c & 1;
    const float* tR1 = sR1[buf];
    const float* tL1 = sL1[buf];
    const float* tR4 = sR4[buf];
    const float* tL4 = sL4[buf];

#if HAVE_TDM
    // retire channel c's 4 tiles; channel c+1's 4 may remain in flight
    if (wid == 0) {
      if (c < NCH - 1) __builtin_amdgcn_s_wait_tensorcnt(4);
      else             __builtin_amdgcn_s_wait_tensorcnt(0);
    }
    __syncthreads();
#else
    copy_tile(sR1[buf], rfeat, lev1, c, rR1, tid);
    copy_tile(sL1[buf], lfeat, lev1, c, rL1, tid);
    copy_tile(sR4[buf], rfeat, lev4, c, rR4, tid);
    copy_tile(sL4[buf], lfeat, lev4, c, rL4, tid);
    __syncthreads();
#endif

    // ---- Phase A: separable bilinear interp from staged texels ----
    for (int idx = tid; idx < 5 * 196; idx += 64) {
      int wy = idx / 196;
      int k  = idx - wy * 196 - 193;
      rv1[idx] = bilerp(tR1, rR1, ix0 + k, iy0 + wy - 2, fxw, fyw);
    }
    for (int idx = tid; idx < 5 * 52; idx += 64) {
      int wy = idx / 52;
      int k  = idx - wy * 52 - 49;
      rv4[idx] = bilerp(tR4, rR4, ix0 + 4 * k, iy0 + 4 * (wy - 2), fxw, fyw);
    }
    if (tid < 25) {
      int wy = tid / 5, wx = tid - wy * 5;
      lv1[tid] = bilerp(tL1, rL1, ix0 + wx - 2,       iy0 + wy - 2,       fxw, fyw);
      lv4[tid] = bilerp(tL4, rL4, ix0 + 4 * (wx - 2), iy0 + 4 * (wy - 2), fxw, fyw);
    }
    __syncthreads();   // texel buffer `buf` free from here on

#if HAVE_TDM
    // refill the just-released buffer with channel c+2 (overlaps Phase B
    // of channel c AND Phase A of channel c+1)
    if (c + 2 < NCH && wid == 0) {
      tdm_load_tile_2d(lds_off_u32(sR1[buf]), tile_src(rfeat, lev1, c + 2, rR1), rR1.cw, rR1.rh);
      tdm_load_tile_2d(lds_off_u32(sL1[buf]), tile_src(lfeat, lev1, c + 2, rL1), rL1.cw, rL1.rh);
      tdm_load_tile_2d(lds_off_u32(sR4[buf]), tile_src(rfeat, lev4, c + 2, rR4), rR4.cw, rR4.rh);
      tdm_load_tile_2d(lds_off_u32(sL4[buf]), tile_src(lfeat, lev4, c + 2, rL4), rL4.cw, rL4.rh);
    }
#endif

    // ---- Phase B: SAD accumulation (3 disparities/lane scale1, 1 scale4) ----
    {
      float lw[25];
#pragma unroll
      for (int i = 0; i < 25; ++i) lw[i] = lv1[i];
#pragma unroll
      for (int ii = 0; ii < 3; ++ii) {
        int d = tid + 64 * ii;
        float a = 0.f;
#pragma unroll
        for (int wy = 0; wy < 5; ++wy) {
          const float* rrow = &rv1[wy * 196 + 191 - d];  // k = wx-2-d -> idx k+193
#pragma unroll
          for (int wx = 0; wx < 5; ++wx)
            a += fabsf(lw[wy * 5 + wx] - rrow[wx]);
        }
        acc1[ii] += a;
      }
    }
    if (tid < ND4) {
      float lw[25];
#pragma unroll
      for (int i = 0; i < 25; ++i) lw[i] = lv4[i];
      int d = tid;
      float a = 0.f;
#pragma unroll
      for (int wy = 0; wy < 5; ++wy) {
        const float* rrow = &rv4[wy * 52 + 47 - d];      // k = wx-2-d -> idx k+49
#pragma unroll
        for (int wx = 0; wx < 5; ++wx)
          a += fabsf(lw[wy * 5 + wx] - rrow[wx]);
      }
      acc4 += a;
    }
    // top-of-loop wait+barrier orders these rv/lv reads against the next
    // channel's Phase A writes
  }

  __syncthreads();
  // ---- finalize: 1-exp(-mean), 48->192 upsample, sum level pair ----
  if (tid < ND4) c4s[tid] = 1.0f - expf(-acc4 * (1.0f / 400.0f));
  __syncthreads();

#pragma unroll
  for (int ii = 0; ii < 3; ++ii) {
    int d = tid + 64 * ii;
    float c1  = 1.0f - expf(-acc1[ii] * (1.0f / 400.0f));
    float pos = (float)d * (47.0f / 191.0f);
    int   i0  = (int)pos;
    int   i1  = (i0 + 1 < ND4) ? i0 + 1 : ND4 - 1;
    float w   = pos - (float)i0;
    float up  = c4s[i0] * (1.0f - w) + c4s[i1] * w;
    out[((size_t)n * MAXD + d) * 8 + lev1] = c1 + up;
  }
}

extern "C" void kernel_launch(void* const* d_in, const int* in_sizes, int n_in,
                              void* d_out, int out_size, void* d_ws, size_t ws_size,
                              hipStream_t stream) {
  (void)in_sizes; (void)n_in; (void)d_ws; (void)ws_size; (void)out_size;
  const float* lfeat = (const float*)d_in[0];
  const float* rfeat = (const float*)d_in[1];
  const float* pts   = (const float*)d_in[2];
  float* out = (float*)d_out;
  // one block per (point, level-pair j): 1024*8 blocks, 64 threads (2 waves)
  PointWiseBlockMaching_26800595927727_kernel<<<NPTS * 8, 64, 0, stream>>>(
      lfeat, rfeat, pts, out);
}